// BottomUpPropagate_50216757624909
// MI455X (gfx1250) — compile-verified
//
#include <hip/hip_runtime.h>
#include <hip/hip_bf16.h>

// Problem constants (fixed by the reference's setup_inputs).
#define BS   32
#define NSEQ 24
#define NB   196
#define NC   196
#define ML   6
#define TI   7          // i-values per block; 28 * 7 == 196 exactly
#define EPSV 1e-6f

typedef __attribute__((ext_vector_type(2))) float v2f;
typedef __attribute__((ext_vector_type(8))) float v8f;

// ---------------------------------------------------------------------------
// Step kernel: for one traversal step, compute upd_raw[b,i] = sub + add*w
// grid = (NB/TI, BS), block = 256 threads (8 waves, wave32).
// ---------------------------------------------------------------------------
__global__ __launch_bounds__(256) void bup_step_kernel(
    const float* __restrict__ ea,        // [BS,NSEQ,NB] current attention (== d_out)
    const float* __restrict__ spo,       // [BS,NSEQ,NB,NC]
    const int*   __restrict__ ctx,       // [BS,NB,NC]
    const int*   __restrict__ roi_cls,   // [BS,NB]
    const float* __restrict__ roi_mask,  // [BS,NB,NC] (exactly 0.0 or 1.0)
    const float* __restrict__ woc,       // [BS,NSEQ,NB]
    const int*   __restrict__ trav,      // [BS,ML]
    const int*   __restrict__ adj,       // [BS,NSEQ,NSEQ]
    float*       __restrict__ ws_upd,    // [BS,NB] workspace
    int step)
{
    __shared__ float s_ea[NSEQ * NB];    // ea[b] pre-scaled by k_cls (18.4 KB)
    __shared__ float s_part[256];        // per-c partials for reduction
    __shared__ int   s_cj[NSEQ];         // compacted child row ids
    __shared__ int   s_ce[NSEQ];         // compacted edge ids
    __shared__ int   s_ncnt;
    __shared__ int   s_pc;

    const int t    = threadIdx.x;
    const int b    = blockIdx.y;
    const int i0   = blockIdx.x * TI;
    const int lane = t & 31;
    const int wid  = t >> 5;

    // Thread 0: parent + compacted child/edge list for this (batch, step).
    if (t == 0) {
        int p   = trav[b * ML + step];
        int pcv = p < 0 ? 0 : p;
        int n   = 0;
        if (p >= 0) {
            for (int j = 0; j < NSEQ; ++j) {
                int e = adj[(b * NSEQ + pcv) * NSEQ + j];
                if (e >= 0) { s_cj[n] = j; s_ce[n] = e; ++n; }
            }
        }
        s_ncnt = n;
        s_pc   = pcv;
    }

    // Stage ea[b] into LDS with k_cls folded in: s_ea[j,k] = ea[b,j,k]*(roi_cls[b,k]!=-1)
    for (int idx = t; idx < NSEQ * NB; idx += 256) {
        int   k  = idx % NB;
        float kc = (roi_cls[b * NB + k] != -1) ? 1.0f : 0.0f;
        s_ea[idx] = ea[(size_t)b * NSEQ * NB + idx] * kc;
    }
    __syncthreads();

    const int   ncnt = s_ncnt;
    const int   pcv  = s_pc;
    const float neff = (ncnt > 0) ? (float)ncnt : 1.0f;

    for (int ii = 0; ii < TI; ++ii) {
        const int i = i0 + ii;

        // Phase 1: thread t == context slot c. Coalesced spo row streaming,
        // ea gather served from LDS. rmask is 0/1 so one multiply covers rm^3.
        float partial = 0.0f;
        if (t < NC) {
            float rm = roi_mask[((size_t)b * NB + i) * NC + t];
            if (rm != 0.0f) {
                int   k   = ctx[((size_t)b * NB + i) * NC + t];
                float acc = 0.0f;
                for (int u = 0; u < ncnt; ++u) {
                    float g = s_ea[s_cj[u] * NB + k];
                    float r = spo[(((size_t)b * NSEQ + s_ce[u]) * NB + i) * NC + t];
                    acc = fmaf(g, r, acc);
                }
                partial = acc * rm;
            }
        }
        s_part[t] = partial;            // t in [196,256) stores 0 padding
        __syncthreads();

        // Phase 2: wave 0 reduces the 256 partials exactly with the matrix
        // pipe: 4 chained V_WMMA_F32_16X16X4_F32, B = ones, C-accumulated.
        // D[m,n] = sum_q sum_k s_part[q*64 + m*4 + k]; fold 16 row-sums after.
        if (wid == 0) {
            float tot;
#if __has_builtin(__builtin_amdgcn_wmma_f32_16x16x4_f32)
            v8f acc8 = {0.f, 0.f, 0.f, 0.f, 0.f, 0.f, 0.f, 0.f};
            v2f bones; bones[0] = 1.0f; bones[1] = 1.0f;
#pragma unroll
            for (int q = 0; q < 4; ++q) {
                const int base = q * 64;
                v2f a;
                // 32-bit A 16x4 layout: lanes 0-15 hold K=0,1; lanes 16-31 hold K=2,3
                if (lane < 16) {
                    a[0] = s_part[base + lane * 4 + 0];
                    a[1] = s_part[base + lane * 4 + 1];
                } else {
                    a[0] = s_part[base + (lane - 16) * 4 + 2];
                    a[1] = s_part[base + (lane - 16) * 4 + 3];
                }
                acc8 = __builtin_amdgcn_wmma_f32_16x16x4_f32(
                    /*neg_a=*/false, a, /*neg_b=*/false, bones,
                    /*c_mod=*/(short)0, acc8, /*reuse_a=*/false, /*reuse_b=*/false);
            }
            // D column 0: rows 0-7 live in lane 0 (vgpr r), rows 8-15 in lane 16.
            float rs = acc8[0] + acc8[1] + acc8[2] + acc8[3] +
                       acc8[4] + acc8[5] + acc8[6] + acc8[7];
            tot = __shfl(rs, 0, 32) + __shfl(rs, 16, 32);
#else
            tot = 0.0f;
            if (lane == 0) for (int q = 0; q < 256; ++q) tot += s_part[q];
#endif
            if (lane == 0) {
                float kci = (roi_cls[b * NB + i] != -1) ? 1.0f : 0.0f;
                float add = kci * tot + neff * EPSV;
                float sub = ea[((size_t)b * NSEQ + pcv) * NB + i];
                float wgt = woc[((size_t)b * NSEQ + pcv) * NB + i];
                ws_upd[b * NB + i] = sub + add * wgt;
            }
        }
        __syncthreads();
    }
}

// ---------------------------------------------------------------------------
// Finalize kernel: per batch, norm by clipped max|upd|, roi mask, row write.
// grid = BS, block = 256.
// ---------------------------------------------------------------------------
__global__ __launch_bounds__(256) void bup_finalize_kernel(
    float*       __restrict__ ea,        // [BS,NSEQ,NB] (== d_out)
    const float* __restrict__ ws_upd,    // [BS,NB]
    const int*   __restrict__ trav,      // [BS,ML]
    const int*   __restrict__ adj,       // [BS,NSEQ,NSEQ]
    const int*   __restrict__ roi_cls,   // [BS,NB]
    int step)
{
    __shared__ float s_red[256];
    __shared__ int   s_write;
    __shared__ int   s_pc;

    const int b = blockIdx.x;
    const int t = threadIdx.x;

    if (t == 0) {
        int p   = trav[b * ML + step];
        int pcv = p < 0 ? 0 : p;
        int n   = 0;
        if (p >= 0)
            for (int j = 0; j < NSEQ; ++j)
                if (adj[(b * NSEQ + pcv) * NSEQ + j] >= 0) ++n;
        s_write = (p >= 0 && n > 0) ? 1 : 0;
        s_pc    = pcv;
    }
    __syncthreads();
    if (!s_write) return;               // row already holds sub_atn: nothing to do

    float u = (t < NB) ? ws_upd[b * NB + t] : 0.0f;
    s_red[t] = fabsf(u);
    __syncthreads();
    for (int off = 128; off > 0; off >>= 1) {
        if (t < off) s_red[t] = fmaxf(s_red[t], s_red[t + off]);
        __syncthreads();
    }
    float norm = s_red[0];
    norm = (norm <= 1.0f) ? 1.0f : norm;

    if (t < NB) {
        float f = (roi_cls[b * NB + t] == -1) ? -1.0f : u / norm;
        ea[((size_t)b * NSEQ + s_pc) * NB + t] = f;
    }
}

// ---------------------------------------------------------------------------
extern "C" void kernel_launch(void* const* d_in, const int* in_sizes, int n_in,
                              void* d_out, int out_size, void* d_ws, size_t ws_size,
                              hipStream_t stream) {
    const int*   trav     = (const int*)  d_in[0];
    const int*   adj      = (const int*)  d_in[1];
    const float* ent_attn = (const float*)d_in[2];
    const float* spo      = (const float*)d_in[3];
    const int*   ctx      = (const int*)  d_in[4];
    const int*   roi_cls  = (const int*)  d_in[5];
    const float* roi_mask = (const float*)d_in[6];
    const float* woc      = (const float*)d_in[7];

    float* ea     = (float*)d_out;       // output: evolved ent_attn [BS,NSEQ,NB]
    float* ws_upd = (float*)d_ws;        // [BS,NB] per-step raw updates

    // Seed output with ent_attn; all further updates are in-place on d_out.
    hipMemcpyAsync(ea, ent_attn, sizeof(float) * BS * NSEQ * NB,
                   hipMemcpyDeviceToDevice, stream);

    for (int step = 0; step < ML; ++step) {
        bup_step_kernel<<<dim3(NB / TI, BS), 256, 0, stream>>>(
            ea, spo, ctx, roi_cls, roi_mask, woc, trav, adj, ws_upd, step);
        bup_finalize_kernel<<<BS, 256, 0, stream>>>(
            ea, ws_upd, trav, adj, roi_cls, step);
    }
}